// Quantizer_24163486007595
// MI455X (gfx1250) — compile-verified
//
#include <hip/hip_runtime.h>

// ---------------------------------------------------------------------------
// Gumbel-softmax vector quantizer, fused forward for MI455X (gfx1250).
//   logits = x @ W^T + b      -> f32 WMMA 16x16x4 (argmax epilogue is
//                                discontinuous => keep full f32 precision)
//   quantized[b,t,g,:] = codebooks[g, argmax(logits+gumbel), :]
//   diversity = mean-entropy of softmax(logits)
// Data movement: double-buffered global->LDS ASYNC DMA (ASYNCcnt) so the
// next K-block's tile transfer overlaps the current block's WMMAs.
// ---------------------------------------------------------------------------

typedef __attribute__((ext_vector_type(2))) float v2f;
typedef __attribute__((ext_vector_type(8))) float v8f;

#define BT    32768   // B*T tokens
#define DDIM  768     // D
#define NOUT  640     // G*V
#define VSZ   320     // V
#define GDIM  384     // D/G
#define KB    16      // K block staged in LDS
#define NKB   (DDIM / KB)   // 48
#define MT    16      // token rows per workgroup
#define NTPW  5       // 16-wide N tiles per wave (5*16 = 80 cols/wave)

__global__ __launch_bounds__(256)
void vq_fused_kernel(const float* __restrict__ x,
                     const float* __restrict__ W,
                     const float* __restrict__ bproj,
                     const float* __restrict__ cb,
                     const float* __restrict__ gum,
                     float* __restrict__ outq,
                     float* __restrict__ gacc)
{
    // Double-buffered tiles. sWL[0] (40KB) is reused as the 16x640 logits
    // tile after the GEMM (disjoint lifetimes).
    __shared__ float sWL[2][NOUT * KB];   // W tiles: [640][16] each
    __shared__ float sA[2][MT * KB];      // x tiles: [16][16] each
    __shared__ float sPacc[NOUT];

    const int tid  = threadIdx.x;
    const int wave = tid >> 5;
    const int lane = tid & 31;
    const int half = lane >> 4;   // 0: lanes 0-15, 1: lanes 16-31
    const int l16  = lane & 15;
    const int row0 = blockIdx.x * MT;

    for (int i = tid; i < NOUT; i += 256) sPacc[i] = 0.0f;

    // LDS byte offsets of the staging buffers (generic ptr low 32 bits ==
    // offset within the wave's LDS allocation).
    const unsigned sW_base = (unsigned)(unsigned long long)&sWL[0][0];
    const unsigned sA_base = (unsigned)(unsigned long long)&sA[0][0];

    // Issue async global->LDS DMA for one K-block into buffer `buf`.
    auto stage = [&](int buf, int k0) {
        if (tid < 64) {  // x tile: 16 rows x 16 K = 64 x b128 (waves 0-1)
            int m = tid >> 2, k4 = (tid & 3) << 2;
            unsigned lds = sA_base + (unsigned)((buf * (MT * KB) + m * KB + k4) * 4);
            unsigned long long ga = (unsigned long long)
                ((const char*)x + ((size_t)(row0 + m) * DDIM + k0 + k4) * 4);
            asm volatile("global_load_async_to_lds_b128 %0, %1, off"
                         :: "v"(lds), "v"(ga) : "memory");
        }
#pragma unroll
        for (int j = 0; j < 10; ++j) {   // W tile: 640 x 16 K = 2560 x b128
            int i = tid + 256 * j;
            int n = i >> 2, k4 = (i & 3) << 2;
            unsigned lds = sW_base + (unsigned)((buf * (NOUT * KB) + n * KB + k4) * 4);
            unsigned long long ga = (unsigned long long)
                ((const char*)W + ((size_t)n * DDIM + k0 + k4) * 4);
            asm volatile("global_load_async_to_lds_b128 %0, %1, off"
                         :: "v"(lds), "v"(ga) : "memory");
        }
    };

    v8f acc[NTPW];
#pragma unroll
    for (int j = 0; j < NTPW; ++j) { v8f z = {}; acc[j] = z; }

    // ---------------- GEMM: logits tile [16 x 640], K = 768 -----------------
    stage(0, 0);
    for (int kb = 0; kb < NKB; ++kb) {
        const int cur = kb & 1;
        // This wave's async loads for `cur` are complete:
        asm volatile("s_wait_asynccnt 0x0" ::: "memory");
        __syncthreads();              // ...and every other wave's too.
        if (kb + 1 < NKB) stage(1 - cur, (kb + 1) * KB);  // overlap next DMA

#pragma unroll
        for (int kk = 0; kk < KB / 4; ++kk) {
            // A fragment 16x4: lanes 0-15 hold K+{0,1}, lanes 16-31 K+{2,3}
            v2f a;
            a.x = sA[cur][l16 * KB + kk * 4 + half * 2 + 0];
            a.y = sA[cur][l16 * KB + kk * 4 + half * 2 + 1];
#pragma unroll
            for (int j = 0; j < NTPW; ++j) {
                int n = wave * 80 + j * 16 + l16;
                v2f b;   // B fragment 4x16 (K x N), symmetric layout
                b.x = sWL[cur][n * KB + kk * 4 + half * 2 + 0];
                b.y = sWL[cur][n * KB + kk * 4 + half * 2 + 1];
                acc[j] = __builtin_amdgcn_wmma_f32_16x16x4_f32(
                    false, a, false, b, (short)0, acc[j], false, false);
            }
        }
        __syncthreads();              // all reads of `cur` done before refill
    }

    // ---------------- bias + spill logits tile to LDS (aliases sWL[0]) ------
    float* Ls = &sWL[0][0];
#pragma unroll
    for (int j = 0; j < NTPW; ++j) {
        int n = wave * 80 + j * 16 + l16;
        float bias = bproj[n];
#pragma unroll
        for (int r = 0; r < 8; ++r) {
            // C/D layout: VGPR r holds row M = r + 8*half, col N = l16 (+tile)
            Ls[(r + 8 * half) * NOUT + n] = acc[j][r] + bias;
        }
    }
    __syncthreads();

    // ---------------- epilogue: 32 (row, group) tasks, 4 per wave ------------
    for (int q = 0; q < 4; ++q) {
        int t   = wave * 4 + q;
        int row = t >> 1;
        int g   = t & 1;
        long gt = row0 + row;

        float lg[10];
#pragma unroll
        for (int i = 0; i < 10; ++i)
            lg[i] = Ls[row * NOUT + g * VSZ + lane + i * 32];

        // max of raw logits (stable softmax)
        float mraw = -3.402823466e+38f;
#pragma unroll
        for (int i = 0; i < 10; ++i) mraw = fmaxf(mraw, lg[i]);
#pragma unroll
        for (int off = 16; off >= 1; off >>= 1)
            mraw = fmaxf(mraw, __shfl_xor(mraw, off, 32));

        // gumbel-perturbed argmax (tau > 0 doesn't change argmax)
        const float* up = gum + ((size_t)gt * 2 + g) * VSZ;
        float zmax = -3.402823466e+38f;
        int   zidx = 0;
#pragma unroll
        for (int i = 0; i < 10; ++i) {
            int v = lane + i * 32;
            float u  = up[v];
            float gn = -__logf(-__logf(u + 1e-10f) + 1e-10f);
            float z  = lg[i] + gn;
            if (z > zmax) { zmax = z; zidx = v; }
        }
#pragma unroll
        for (int off = 16; off >= 1; off >>= 1) {
            float oz = __shfl_xor(zmax, off, 32);
            int   oi = __shfl_xor(zidx, off, 32);
            if (oz > zmax || (oz == zmax && oi < zidx)) { zmax = oz; zidx = oi; }
        }

        // softmax of raw logits -> per-WG partial p_bar accumulation (LDS)
        float se = 0.0f;
#pragma unroll
        for (int i = 0; i < 10; ++i) se += __expf(lg[i] - mraw);
#pragma unroll
        for (int off = 16; off >= 1; off >>= 1) se += __shfl_xor(se, off, 32);
        float inv_se = 1.0f / se;
#pragma unroll
        for (int i = 0; i < 10; ++i) {
            float p = __expf(lg[i] - mraw) * inv_se;
            atomicAdd(&sPacc[g * VSZ + lane + i * 32], p);   // ds_add_f32
        }

        // one-hot codebook gather: 384 floats = 96 float4, 3 per lane
        const float4* cbp = (const float4*)(cb + ((size_t)g * VSZ + zidx) * GDIM);
        float4*       op  = (float4*)(outq + (size_t)gt * DDIM + g * GDIM);
#pragma unroll
        for (int i = 0; i < 3; ++i) op[lane + i * 32] = cbp[lane + i * 32];
    }

    // fold per-WG partial softmax sums into global accumulator
    __syncthreads();
    for (int i = tid; i < NOUT; i += 256) atomicAdd(&gacc[i], sPacc[i]);
}

__global__ void vq_zero_acc(float* __restrict__ gacc) {
    int i = blockIdx.x * blockDim.x + threadIdx.x;
    if (i < NOUT) gacc[i] = 0.0f;
}

__global__ __launch_bounds__(256)
void vq_finalize(const float* __restrict__ gacc, float* __restrict__ loss_out) {
    __shared__ float red[256];
    int tid = threadIdx.x;
    float s = 0.0f;
    for (int i = tid; i < NOUT; i += 256) {
        float pb = gacc[i] * (1.0f / (float)BT);
        s += pb * __logf(pb + 1e-9f);
    }
    red[tid] = s;
    __syncthreads();
    for (int stride = 128; stride > 0; stride >>= 1) {
        if (tid < stride) red[tid] += red[tid + stride];
        __syncthreads();
    }
    if (tid == 0) loss_out[0] = red[0] * (1.0f / (float)NOUT);
}

extern "C" void kernel_launch(void* const* d_in, const int* in_sizes, int n_in,
                              void* d_out, int out_size, void* d_ws, size_t ws_size,
                              hipStream_t stream) {
    (void)in_sizes; (void)n_in; (void)out_size; (void)ws_size;
    const float* x   = (const float*)d_in[0];   // [B,T,D]
    const float* W   = (const float*)d_in[1];   // [G*V, D]
    const float* b   = (const float*)d_in[2];   // [G*V]
    const float* cb  = (const float*)d_in[3];   // [G,V,GD]
    const float* gum = (const float*)d_in[4];   // [B,T,G,V]

    float* outq = (float*)d_out;                     // quantized [B,T,D]
    float* loss = outq + (size_t)BT * DDIM;          // scalar after it
    float* gacc = (float*)d_ws;                      // 640 floats of scratch

    vq_zero_acc<<<2, 320, 0, stream>>>(gacc);
    vq_fused_kernel<<<BT / MT, 256, 0, stream>>>(x, W, b, cb, gum, outq, gacc);
    vq_finalize<<<1, 256, 0, stream>>>(gacc, loss);
}